// GNNLayer_7516192768269
// MI455X (gfx1250) — compile-verified
//
#include <hip/hip_runtime.h>

#define N_NODES 100000
#define N_EDGES 1600000
#define F 48          // IN_FEATS == OUT_FEATS == 48

typedef __attribute__((ext_vector_type(2))) float v2f;
typedef __attribute__((ext_vector_type(8))) float v8f;

// ---------------------------------------------------------------------------
// Kernel 0: zero the workspace (sums [N*F] followed by deg [N])
// ---------------------------------------------------------------------------
__global__ void gnn_zero_ws(float* __restrict__ ws, int n) {
    int i = blockIdx.x * blockDim.x + threadIdx.x;
    if (i < n) ws[i] = 0.0f;
}

// ---------------------------------------------------------------------------
// Kernel 1: edge-parallel scatter-add.  One thread per (edge, feature).
// feature (19.2 MB) and sums (19.2 MB) both live in the 192 MB L2, so these
// fp32 atomics resolve at the L2 atomic units (GLOBAL_ATOMIC_ADD_F32,
// no-return -> STOREcnt).  48 consecutive threads share an edge, so the
// gather of the 192-byte feature row is coalesced.
// ---------------------------------------------------------------------------
__global__ void gnn_scatter_accum(const float* __restrict__ feat,
                                  const int*   __restrict__ src,
                                  const int*   __restrict__ dst,
                                  float* __restrict__ sums,
                                  float* __restrict__ deg) {
    long idx   = (long)blockIdx.x * blockDim.x + threadIdx.x;
    long total = (long)N_EDGES * F;
    if (idx >= total) return;
    int e = (int)(idx / F);
    int f = (int)(idx % F);
    int s = src[e];
    int d = dst[e];
    float v = feat[(long)s * F + f];
    unsafeAtomicAdd(&sums[(long)d * F + f], v);       // global_atomic_add_f32
    if (f == 0) unsafeAtomicAdd(&deg[d], 1.0f);
}

// ---------------------------------------------------------------------------
// Kernel 2: mean + linear via V_WMMA_F32_16X16X4_F32 (exact fp32 path).
// One wave32 per 16-node tile.  D[m][n] = sum_k mean[m][k] * W[n][k] + b[n].
// A = mean tile (16x4 per step), B = W^T tile (4x16), C/D = 16x16 f32.
//
// Fragment layouts (ISA 7.12.2, wave32):
//   A 16x4 f32 : lanes 0-15 -> M=lane, v0=K0, v1=K1 ; lanes 16-31 -> K2/K3
//   B 4x16 f32 : lanes 0-15 -> N=lane, v0=K0, v1=K1 ; lanes 16-31 -> K2/K3
//   C/D 16x16  : vgpr r -> M=r (lanes 0-15) / M=r+8 (lanes 16-31), N=lane&15
// ---------------------------------------------------------------------------
__global__ void __launch_bounds__(32)
gnn_mean_linear_wmma(const float* __restrict__ sums,
                     const float* __restrict__ deg,
                     const float* __restrict__ W,     // [F][F], row n = out feat
                     const float* __restrict__ bias,  // [F]
                     float* __restrict__ out) {       // [N][F]
    __shared__ float Wl[F * F];
    __shared__ float bl[F];

    const int lane = threadIdx.x;          // 0..31
    for (int i = lane; i < F * F; i += 32) Wl[i] = W[i];
    if (lane < F) bl[lane] = bias[lane];
    __syncthreads();

    const int  tile   = blockIdx.x;        // 6250 tiles of 16 nodes (exact)
    const int  row0   = tile * 16;
    const int  m      = lane & 15;         // A-row / D-col within tile
    const bool hiHalf = lane >= 16;        // lanes 16-31 hold K+2/K+3, M+8

    const int   row  = row0 + m;
    const float dsc  = 1.0f / fmaxf(deg[row], 1.0f);
    const float* srow = sums + (long)row * F + (hiHalf ? 2 : 0);

    v8f c0 = {}; v8f c1 = {}; v8f c2 = {};

#pragma unroll
    for (int kc = 0; kc < F / 4; ++kc) {   // 12 K-steps of 4
        // A fragment: two consecutive mean values (8B load), scaled by 1/deg
        v2f a = *(const v2f*)(srow + kc * 4);
        a *= dsc;
        // B fragments for the three 16-wide N tiles: W[n][kb], W[n][kb+1]
        const int kb = kc * 4 + (hiHalf ? 2 : 0);
        v2f b0 = *(const v2f*)&Wl[( 0 + m) * F + kb];
        v2f b1 = *(const v2f*)&Wl[(16 + m) * F + kb];
        v2f b2 = *(const v2f*)&Wl[(32 + m) * F + kb];
        // 8 args: (neg_a, A, neg_b, B, c_mod, C, reuse_a, reuse_b)
        c0 = __builtin_amdgcn_wmma_f32_16x16x4_f32(false, a, false, b0,
                                                   (short)0, c0, false, false);
        c1 = __builtin_amdgcn_wmma_f32_16x16x4_f32(false, a, false, b1,
                                                   (short)0, c1, false, false);
        c2 = __builtin_amdgcn_wmma_f32_16x16x4_f32(false, a, false, b2,
                                                   (short)0, c2, false, false);
    }

    // Write D + bias.  Lanes 0-15 cover rows r, lanes 16-31 rows r+8; each
    // 32-lane store covers two rows' 64-byte segments.
    const int mbase = hiHalf ? 8 : 0;
#pragma unroll
    for (int r = 0; r < 8; ++r) {
        const long orow = (long)(row0 + mbase + r) * F;
        out[orow +  0 + m] = c0[r] + bl[ 0 + m];
        out[orow + 16 + m] = c1[r] + bl[16 + m];
        out[orow + 32 + m] = c2[r] + bl[32 + m];
    }
}

// ---------------------------------------------------------------------------
// Launch: inputs in setup_inputs() order: feature, W, b, src, dst
// ---------------------------------------------------------------------------
extern "C" void kernel_launch(void* const* d_in, const int* in_sizes, int n_in,
                              void* d_out, int out_size, void* d_ws, size_t ws_size,
                              hipStream_t stream) {
    const float* feat = (const float*)d_in[0];
    const float* W    = (const float*)d_in[1];
    const float* b    = (const float*)d_in[2];
    const int*   src  = (const int*)d_in[3];
    const int*   dst  = (const int*)d_in[4];
    float*       out  = (float*)d_out;

    float* sums = (float*)d_ws;                    // N*F floats
    float* deg  = sums + (size_t)N_NODES * F;      // N floats

    // 1) zero sums+deg
    const int ztotal = N_NODES * F + N_NODES;
    gnn_zero_ws<<<(ztotal + 255) / 256, 256, 0, stream>>>(sums, ztotal);

    // 2) scatter-add messages + degree
    const long stotal = (long)N_EDGES * F;
    gnn_scatter_accum<<<(unsigned)((stotal + 255) / 256), 256, 0, stream>>>(
        feat, src, dst, sums, deg);

    // 3) mean + linear with f32 WMMA, one wave per 16-node tile
    gnn_mean_linear_wmma<<<N_NODES / 16, 32, 0, stream>>>(sums, deg, W, b, out);
}